// DirectionVarEntropy_24902220382276
// MI455X (gfx1250) — compile-verified
//
#include <hip/hip_runtime.h>
#include <cstdint>
#include <cstddef>

#define PH    14
#define NWIN  12      // PH - 3 + 1
#define BINS  256
#define WAVES 4       // waves (sites) per block

// population std of 3 values, two-pass like numpy; raw v_sqrt_f32 (~1 ulp)
__device__ __forceinline__ float std3(float a, float b, float c) {
    float m  = (a + b + c) * (1.0f / 3.0f);
    float da = a - m, db = b - m, dc = c - m;
    return __builtin_amdgcn_sqrtf((da * da + db * db + dc * dc) * (1.0f / 3.0f));
}

__device__ __forceinline__ float waveSum(float v) {
#pragma unroll
    for (int off = 16; off > 0; off >>= 1) v += __shfl_xor(v, off, 32);
    return v;
}

// psi for one 3x3 window at flat index w inside the 14x14 patch P0
__device__ __forceinline__ float window_psi(const float* __restrict__ P0, int w,
                                            float c0, float c1, float c2) {
    int wi = w / NWIN;
    int wj = w - wi * NWIN;
    const float* P = P0 + wi * PH + wj;
    float w00 = P[0],  w01 = P[1],  w02 = P[2];
    float w10 = P[14], w11 = P[15], w12 = P[16];
    float w20 = P[28], w21 = P[29], w22 = P[30];

    // vertical pass: t[i][k] = sum_j D[i][j] * W[j][k]
    float t00 = c0 * (w00 + w10 + w20);
    float t01 = c0 * (w01 + w11 + w21);
    float t02 = c0 * (w02 + w12 + w22);
    float t10 = c1 * (w00 - w20);
    float t11 = c1 * (w01 - w21);
    float t12 = c1 * (w02 - w22);
    float t20 = c2 * (w00 - 2.0f * w10 + w20);
    float t21 = c2 * (w01 - 2.0f * w11 + w21);
    float t22 = c2 * (w02 - 2.0f * w12 + w22);

    // horizontal pass: d[i][l] = sum_k t[i][k] * D[l][k]
    float d00 = c0 * (t00 + t01 + t02);
    float d01 = c1 * (t00 - t02);
    float d02 = c2 * (t00 - 2.0f * t01 + t02);
    float d10 = c0 * (t10 + t11 + t12);
    float d11 = c1 * (t10 - t12);
    float d12 = c2 * (t10 - 2.0f * t11 + t12);
    float d20 = c0 * (t20 + t21 + t22);
    float d21 = c1 * (t20 - t22);
    float d22 = c2 * (t20 - 2.0f * t21 + t22);

    float S1 = (std3(d00, d01, d02) + std3(d10, d11, d12) + std3(d20, d21, d22)) * (1.0f / 3.0f);
    float S2 = (std3(d00, d10, d20) + std3(d01, d11, d21) + std3(d02, d12, d22)) * (1.0f / 3.0f);
    float S3 = std3(d00, d11, d22);
    float S4 = std3(d02, d11, d20);   // diag of column-flipped block

    // psi = var({Si/S}, ddof=1) == var({Si}, ddof=1) / S^2
    float mu = (S1 + S2 + S3 + S4) * 0.25f;
    float S  = mu + 1e-8f;
    float rS = __builtin_amdgcn_rcpf(S);     // v_rcp_f32, 1 ulp
    float e1 = S1 - mu, e2 = S2 - mu, e3 = S3 - mu, e4 = S4 - mu;
    float devsum = e1 * e1 + e2 * e2 + e3 * e3 + e4 * e4;
    return devsum * (1.0f / 3.0f) * rS * rS;
}

__global__ __launch_bounds__(WAVES * 32)
void richness_kernel(const float* __restrict__ x, const float* __restrict__ dctm,
                     float* __restrict__ out, int nsites) {
    __shared__ float    patch[WAVES][PH * PH];
    __shared__ unsigned hist[WAVES][BINS];

    const int lane = threadIdx.x & 31;
    const int wv   = threadIdx.x >> 5;
    const int site = blockIdx.x * WAVES + wv;
    if (site >= nsites) return;

    const int b   = site >> 10;        // Hp*Wp = 1024
    const int rem = site & 1023;
    const int hp  = rem >> 5;          // Wp = 32
    const int wp  = rem & 31;

    // exact DCT-matrix entries from the reference input:
    // row0 = c0*(1,1,1), row1 = c1*(1,0,-1), row2 = c2*(1,-2,1)
    const float c0 = dctm[0];
    const float c1 = dctm[3];
    const float c2 = dctm[6];

    float res = 0.0f;

    for (int ch = 0; ch < 3; ++ch) {
        const float* gbase =
            x + (((size_t)(b * 3 + ch) * 448) + (size_t)hp * PH) * 448 + (size_t)wp * PH;

        // make sure previous channel's DS reads retired before LDS is overwritten
        asm volatile("s_wait_dscnt 0" ::: "memory");

        // ---- async-stage the 14x14 patch into this wave's LDS slab ----
#pragma unroll
        for (int r = 0; r < 7; ++r) {
            int idx  = lane + 32 * r;
            int cidx = idx > PH * PH - 1 ? PH * PH - 1 : idx;   // clamp: dup writes, same data
            int row  = cidx / PH;
            int col  = cidx - row * PH;
            const float* gp = gbase + row * 448 + col;
            unsigned loff = (unsigned)(uintptr_t)(&patch[wv][cidx]);
            asm volatile("global_load_async_to_lds_b32 %0, %1, off"
                         :: "v"(loff), "v"((unsigned long long)(uintptr_t)gp)
                         : "memory");
        }

        // zero this wave's histogram while the async copies are in flight
#pragma unroll
        for (int t = 0; t < 8; ++t) hist[wv][lane + 32 * t] = 0u;

        asm volatile("s_wait_asynccnt 0" ::: "memory");

        // ---- histogram via LDS atomics (wave-private region, DS in-order per wave) ----
#pragma unroll
        for (int r = 0; r < 7; ++r) {
            int idx = lane + 32 * r;
            if (idx < PH * PH) {
                float v  = patch[wv][idx];
                int  bin = (int)rintf(v * 255.0f);       // round-half-even == jnp.round
                bin = bin < 0 ? 0 : (bin > 255 ? 255 : bin);
                atomicAdd(&hist[wv][bin], 1u);
            }
        }

        // ---- entropy: 8 bins per lane, then wave reduction ----
        float esum = 0.0f;
#pragma unroll
        for (int t = 0; t < 8; ++t) {
            float cnt = (float)hist[wv][lane + 32 * t];
            float p   = fmaf(cnt, 1.0f / 196.0f, 1e-10f);
            esum     -= p * __builtin_amdgcn_logf(p);    // v_log_f32 == log2
        }
        esum = waveSum(esum);

        // ---- 144 sliding 3x3 DCT windows: 4 guard-free per lane + guarded tail ----
        const float* P0 = &patch[wv][0];
        float psum = 0.0f;
#pragma unroll
        for (int k = 0; k < 4; ++k)                       // w = lane + 32k <= 127 < 144
            psum += window_psi(P0, lane + 32 * k, c0, c1, c2);
        if (lane < 16)                                    // w = lane + 128 < 144
            psum += window_psi(P0, lane + 128, c0, c1, c2);

        psum = waveSum(psum);
        float psi_m = psum * (1.0f / 144.0f);

        res += psi_m * esum;
    }

    if (lane == 0) out[site] = res / 3.0f;
}

extern "C" void kernel_launch(void* const* d_in, const int* in_sizes, int n_in,
                              void* d_out, int out_size, void* d_ws, size_t ws_size,
                              hipStream_t stream) {
    (void)in_sizes; (void)n_in; (void)d_ws; (void)ws_size;
    const float* x    = (const float*)d_in[0];
    const float* dctm = (const float*)d_in[1];
    float*       out  = (float*)d_out;

    int nsites = out_size;                       // 16 * 32 * 32 = 16384
    int blocks = (nsites + WAVES - 1) / WAVES;   // 4096 blocks of 128 threads
    richness_kernel<<<blocks, WAVES * 32, 0, stream>>>(x, dctm, out, nsites);
}